// DensetntPostprocess_82884278879132
// MI455X (gfx1250) — compile-verified
//
#include <hip/hip_runtime.h>
#include <stdint.h>

#define B_SZ      128
#define N_CAND    4096
#define PSTEPS    30
#define MODE_NUM  6
#define THR2      4.0f
#define NTHREADS  256
#define NWAVES    (NTHREADS / 32)
#define PER_THREAD 16            // N_CAND / NTHREADS

// CDNA5 async global->LDS copy (ASYNCcnt-tracked), GV addressing mode.
// %0 = VGPR holding LDS byte address (low 32 bits of generic shared ptr),
// %1 = VGPR pair holding 64-bit global address.
__device__ __forceinline__ void async_f4_to_lds(uint32_t lds_byte_addr, const float* gptr) {
    uint64_t ga = (uint64_t)(uintptr_t)gptr;
    asm volatile("global_load_async_to_lds_b128 %0, %1, off"
                 :: "v"(lds_byte_addr), "v"(ga) : "memory");
}

__device__ __forceinline__ void wait_asynccnt0() {
#if __has_builtin(__builtin_amdgcn_s_wait_asynccnt)
    __builtin_amdgcn_s_wait_asynccnt(0);
#else
    asm volatile("s_wait_asynccnt 0" ::: "memory");
#endif
}

__global__ __launch_bounds__(NTHREADS)
void densetnt_nms_kernel(const float* __restrict__ scores,  // [B, N]
                         const float* __restrict__ trajs,   // [B, 2P, N]
                         const float* __restrict__ goals,   // [B, 2, N]
                         float* __restrict__ out)           // B*6*60 trajs ++ B*6 scores
{
    __shared__ __align__(16) float sS [N_CAND];
    __shared__ __align__(16) float sGx[N_CAND];
    __shared__ __align__(16) float sGy[N_CAND];
    __shared__ float sWaveS[NWAVES];
    __shared__ int   sWaveI[NWAVES];
    __shared__ int   sSelIdx[MODE_NUM];
    __shared__ float sSelScore[MODE_NUM];
    __shared__ int   sFound;

    const int tid = threadIdx.x;
    const int b   = blockIdx.x;

    const float* rowS  = scores + (size_t)b * N_CAND;
    const float* rowGx = goals  + (size_t)b * 2 * N_CAND;
    const float* rowGy = rowGx + N_CAND;
    const float* rowT  = trajs  + (size_t)b * (2 * PSTEPS) * N_CAND;

    const uint32_t ldsS  = (uint32_t)(uintptr_t)(&sS[0]);
    const uint32_t ldsGx = (uint32_t)(uintptr_t)(&sGx[0]);
    const uint32_t ldsGy = (uint32_t)(uintptr_t)(&sGy[0]);

    // ---- Phase 1: async-stage scores + goal coords into LDS (48 KB) ----
    #pragma unroll
    for (int j = 0; j < 4; ++j) {
        const int f4 = tid + j * NTHREADS;            // float4 index within row
        const uint32_t off16 = (uint32_t)f4 * 16u;
        async_f4_to_lds(ldsS  + off16, rowS  + (size_t)f4 * 4);
        async_f4_to_lds(ldsGx + off16, rowGx + (size_t)f4 * 4);
        async_f4_to_lds(ldsGy + off16, rowGy + (size_t)f4 * 4);
    }
    wait_asynccnt0();
    __syncthreads();

    // ---- Phase 2: pull this thread's 16 candidates into registers ----
    // Candidate n = tid*4 + (c>>2)*1024 + (c&3)  for register slot c.
    float rS[PER_THREAD], rGx[PER_THREAD], rGy[PER_THREAD];
    #pragma unroll
    for (int j = 0; j < 4; ++j) {
        const float4 a = ((const float4*)sS )[tid + j * NTHREADS];
        const float4 x = ((const float4*)sGx)[tid + j * NTHREADS];
        const float4 y = ((const float4*)sGy)[tid + j * NTHREADS];
        rS [j*4+0]=a.x; rS [j*4+1]=a.y; rS [j*4+2]=a.z; rS [j*4+3]=a.w;
        rGx[j*4+0]=x.x; rGx[j*4+1]=x.y; rGx[j*4+2]=x.z; rGx[j*4+3]=x.w;
        rGy[j*4+0]=y.x; rGy[j*4+1]=y.y; rGy[j*4+2]=y.z; rGy[j*4+3]=y.w;
    }

    // ---- Phase 3: 6 masked-argmax passes (== greedy NMS in score order) ----
    uint32_t supp = 0u;
    for (int pass = 0; pass < MODE_NUM; ++pass) {
        float bs = -3.402823466e38f;       // -FLT_MAX sentinel
        int   bi = 0x7FFFFFFF;
        #pragma unroll
        for (int c = 0; c < PER_THREAD; ++c) {
            if (((supp >> c) & 1u) == 0u) {
                const float s = rS[c];
                const int n = (tid << 2) + ((c >> 2) << 10) + (c & 3);
                if (s > bs || (s == bs && n < bi)) { bs = s; bi = n; }
            }
        }
        // wave32 pair-reduction (higher score wins; ties -> lower index)
        #pragma unroll
        for (int off = 16; off > 0; off >>= 1) {
            const float os = __shfl_down(bs, off, 32);
            const int   oi = __shfl_down(bi, off, 32);
            if (os > bs || (os == bs && oi < bi)) { bs = os; bi = oi; }
        }
        if ((tid & 31) == 0) { sWaveS[tid >> 5] = bs; sWaveI[tid >> 5] = bi; }
        __syncthreads();
        if (tid == 0) {
            float fs = sWaveS[0]; int fi = sWaveI[0];
            #pragma unroll
            for (int w = 1; w < NWAVES; ++w) {
                const float os = sWaveS[w]; const int oi = sWaveI[w];
                if (os > fs || (os == fs && oi < fi)) { fs = os; fi = oi; }
            }
            const int found = (fi != 0x7FFFFFFF) ? 1 : 0;
            if (!found) { fi = sSelIdx[0]; fs = sSelScore[0]; }   // fallback: top candidate
            sSelIdx[pass] = fi; sSelScore[pass] = fs; sFound = found;
        }
        __syncthreads();
        if (sFound) {
            const int sel = sSelIdx[pass];
            const float sx = sGx[sel], sy = sGy[sel];   // LDS broadcast read
            #pragma unroll
            for (int c = 0; c < PER_THREAD; ++c) {
                const float dx = rGx[c] - sx;
                const float dy = rGy[c] - sy;
                if (dx * dx + dy * dy < THR2) supp |= (1u << c);   // strict <, self-suppressing
            }
        }
    }
    __syncthreads();

    // ---- Phase 4: gather outputs ----
    float* outTraj  = out;                                            // [B,6,30,2]
    float* outScore = out + (size_t)B_SZ * MODE_NUM * PSTEPS * 2;     // [B,6]
    for (int i = tid; i < MODE_NUM * PSTEPS * 2; i += NTHREADS) {
        const int m = i / (PSTEPS * 2);
        const int k = i % (PSTEPS * 2);
        const int n = sSelIdx[m];
        outTraj[(size_t)b * (MODE_NUM * PSTEPS * 2) + i] = rowT[(size_t)k * N_CAND + n];
    }
    if (tid < MODE_NUM) outScore[b * MODE_NUM + tid] = sSelScore[tid];
}

extern "C" void kernel_launch(void* const* d_in, const int* in_sizes, int n_in,
                              void* d_out, int out_size, void* d_ws, size_t ws_size,
                              hipStream_t stream) {
    (void)in_sizes; (void)n_in; (void)d_ws; (void)ws_size; (void)out_size;
    const float* scores = (const float*)d_in[0];   // goals_scores [128, 4096]
    const float* trajs  = (const float*)d_in[1];   // traj_preds  [128, 60, 4096]
    const float* goals  = (const float*)d_in[2];   // pred_goals  [128, 2, 4096]
    float* out = (float*)d_out;
    densetnt_nms_kernel<<<B_SZ, NTHREADS, 0, stream>>>(scores, trajs, goals, out);
}